// _ScaledDotProductAttention_24489903522711
// MI455X (gfx1250) — compile-verified
//
#include <hip/hip_runtime.h>

typedef __attribute__((ext_vector_type(2))) float v2f;
typedef __attribute__((ext_vector_type(4))) float v4f;
typedef __attribute__((ext_vector_type(8))) float v8f;

constexpr int kBS  = 4;
constexpr int kH   = 16;
constexpr int kS   = 1024;
constexpr int kD   = 64;
constexpr float kScale = 0.125f;            // 64^-0.5
constexpr float kNegInf = -1.0e30f;

// XOR swizzle: kills the 16-way bank conflict on row-major 16x1024 LDS tile.
// Only touches col bits [5:2] -> even/odd pairs stay adjacent.
__device__ __forceinline__ int sidx(int m, int c) {
    return m * kS + (c ^ (m << 2));
}

__global__ __launch_bounds__(128)
void attn_fused_kernel(const float* __restrict__ q,
                       const float* __restrict__ k,     // (B,H,D,S) pre-transposed
                       const float* __restrict__ v,     // (B,H,S,D)
                       const float* __restrict__ prev,  // (B,H,S,S)
                       float* __restrict__ out_g,       // (B,H,S,D)
                       float* __restrict__ attn_g,      // (B,H,S,S)
                       float* __restrict__ scores_g)    // (B,H,S,S)
{
    __shared__ float s_scores[16 * kS];     // 64 KB: one 16-row score/prob strip

    const int bh   = blockIdx.x >> 6;       // b*H + h
    const int qt   = blockIdx.x & 63;       // query tile index (16 rows)
    const int tid  = threadIdx.x;
    const int lane = tid & 31;
    const int wave = tid >> 5;              // 0..3
    const int m    = lane & 15;             // row-in-tile / col-in-tile lane id
    const int hi   = lane >> 4;             // 0 or 1 (half-wave select)

    const int NT = qt + 1;                  // causal: key tiles 0..qt live
    const int L  = NT * 16;

    const size_t qkv_base = (size_t)bh * kS * kD;
    const size_t mat_base = (size_t)bh * kS * kS;
    const float* q_head    = q + qkv_base;
    const float* k_head    = k + qkv_base;
    const float* v_head    = v + qkv_base;
    const float* prev_head = prev + mat_base;
    float* scores_h = scores_g + mat_base;
    float* attn_h   = attn_g + mat_base;

    // ---- Q-tile A fragments, ISA layout: lane=M, K = vgpr + 2*hi (+4 per chunk)
    const v2f* qrow2 = (const v2f*)(q_head + (size_t)(qt * 16 + m) * kD);
    v2f qa[16];
#pragma unroll
    for (int kk = 0; kk < 16; ++kk) qa[kk] = qrow2[2 * kk + hi];

    // ---- phase 1: scores tiles = (Q Kt)*scale + prev, causal mask ----
    for (int j0 = wave; j0 < NT; j0 += 4) {
        const float* kb = k_head + (size_t)(j0 * 16 + m);   // B frag: k[d][s0+m]
        v8f acc = {};
#pragma unroll
        for (int kk = 0; kk < 16; ++kk) {
            const int kd = 4 * kk + 2 * hi;
            v2f b;
            b.x = kb[(size_t)kd * kS];
            b.y = kb[(size_t)(kd + 1) * kS];
            acc = __builtin_amdgcn_wmma_f32_16x16x4_f32(false, qa[kk], false, b,
                                                        (short)0, acc, false, false);
        }
        const int col = j0 * 16 + m;
#pragma unroll
        for (int vv = 0; vv < 8; ++vv) {
            const int mm  = vv + 8 * hi;            // C/D layout: M = vgpr + 8*hi
            const int row = qt * 16 + mm;
            float val = acc[vv] * kScale + prev_head[(size_t)row * kS + col];
            if (col > row) val = kNegInf;
            s_scores[sidx(mm, col)] = val;
            scores_h[(size_t)row * kS + col] = val;
        }
    }

    // ---- fully-masked tail tiles: scores=-1e30, attn=0 (no compute) ----
    {
        const int W = kS - L;                       // multiple of 16
        if (W > 0) {
            const int w4 = W >> 2;
            const v4f nf = {kNegInf, kNegInf, kNegInf, kNegInf};
            const v4f zf = {0.f, 0.f, 0.f, 0.f};
            for (int idx = tid; idx < 16 * w4; idx += 128) {
                const int rr = idx / w4;
                const int cc = idx - rr * w4;
                const size_t off = (size_t)(qt * 16 + rr) * kS + L + 4 * cc;
                *(v4f*)(scores_h + off) = nf;
                *(v4f*)(attn_h + off)   = zf;
            }
        }
    }

    __syncthreads();

    // ---- phase 2: row softmax over [0, L); masked -1e30 underflows to 0 ----
    {
        const int r  = tid >> 3;                    // 0..15, 8 lanes per row
        const int c8 = tid & 7;
        float mx = -3.0e38f;
        for (int c = c8; c < L; c += 8) mx = fmaxf(mx, s_scores[sidx(r, c)]);
#pragma unroll
        for (int off = 4; off > 0; off >>= 1) mx = fmaxf(mx, __shfl_xor(mx, off, 8));
        float sum = 0.f;
        for (int c = c8; c < L; c += 8) sum += __expf(s_scores[sidx(r, c)] - mx);
#pragma unroll
        for (int off = 4; off > 0; off >>= 1) sum += __shfl_xor(sum, off, 8);
        const float inv = 1.0f / sum;
        const size_t rowoff = (size_t)(qt * 16 + r) * kS;
        for (int c = c8; c < L; c += 8) {
            const float p = __expf(s_scores[sidx(r, c)] - mx) * inv;
            s_scores[sidx(r, c)] = p;
            attn_h[rowoff + c]   = p;
        }
    }

    __syncthreads();

    // ---- phase 3: out = attn @ V, one 16-wide D tile per wave ----
    {
        const int d0 = wave * 16;
        const float* vb = v_head + d0 + m;          // B frag: v[j][d0+m]
        v8f oacc = {};
        for (int j0 = 0; j0 < L; j0 += 4) {
            const int jj = j0 + 2 * hi;
            v2f a, b;
            a.x = s_scores[sidx(m, jj)];
            a.y = s_scores[sidx(m, jj + 1)];
            b.x = vb[(size_t)jj * kD];
            b.y = vb[(size_t)(jj + 1) * kD];
            oacc = __builtin_amdgcn_wmma_f32_16x16x4_f32(false, a, false, b,
                                                         (short)0, oacc, false, false);
        }
        float* ob = out_g + qkv_base + d0 + m;
#pragma unroll
        for (int vv = 0; vv < 8; ++vv)
            ob[(size_t)(qt * 16 + vv + 8 * hi) * kD] = oacc[vv];
    }
}

extern "C" void kernel_launch(void* const* d_in, const int* in_sizes, int n_in,
                              void* d_out, int out_size, void* d_ws, size_t ws_size,
                              hipStream_t stream) {
    (void)in_sizes; (void)n_in; (void)out_size; (void)d_ws; (void)ws_size;
    const float* q    = (const float*)d_in[0];
    const float* k    = (const float*)d_in[1];
    const float* v    = (const float*)d_in[2];
    const float* prev = (const float*)d_in[3];

    float* out    = (float*)d_out;
    float* attn   = out + (size_t)kBS * kH * kS * kD;
    float* scores = attn + (size_t)kBS * kH * kS * kS;

    const dim3 grid(kBS * kH * (kS / 16));   // 4096 blocks, one per (head, q-tile)
    attn_fused_kernel<<<grid, 128, 0, stream>>>(q, k, v, prev, out, attn, scores);
}